// MF_9612136808809
// MI455X (gfx1250) — compile-verified
//
#include <hip/hip_runtime.h>

typedef __attribute__((ext_vector_type(2))) float v2f;
typedef __attribute__((ext_vector_type(8))) float v8f;

#define K_DIM 64
#define TILE  16

// One wave computes one 16x16 tile of pref = sel @ item_emb.T using
// V_WMMA_F32_16X16X4_F32 (K stepped by 4, 16 steps for K=64).
// Also accumulates sum(s^2) into *loss.
__global__ __launch_bounds__(256) void mf_gemm_wmma(
    const float* __restrict__ user_emb,
    const float* __restrict__ item_emb,
    const int*   __restrict__ center_uid,
    float* __restrict__ pref,    // [B, N]
    float* __restrict__ loss,
    int B, int N, int NT, int totalTiles)
{
  __shared__ float red[8];
  const int waveId = threadIdx.x >> 5;
  const int lane   = threadIdx.x & 31;
  const int wid    = blockIdx.x * (blockDim.x >> 5) + waveId;

  float lsum = 0.0f;
  if (wid < totalTiles) {                    // wave-uniform guard: EXEC stays all-1s
    const int nTile = wid % NT;              // consecutive waves -> consecutive N tiles
    const int mTile = wid / NT;
    const int half  = lane >> 4;             // 0: lanes 0-15, 1: lanes 16-31
    const int ml    = lane & 15;

    // A fragment: row m = ml of gathered user block; k = k0 + 2*half + {0,1}
    int mIdx = mTile * TILE + ml;
    if (mIdx >= B) mIdx = B - 1;             // safety clamp (B%16==0 here)
    const int uid = center_uid[mIdx];
    const float* aPtr = user_emb + (size_t)uid * K_DIM + 2 * half;

    // B fragment: B[k][n] = item_emb[n][k]; n = ml, k = k0 + 2*half + {0,1}
    int nIdx = nTile * TILE + ml;
    const int nClamped = (nIdx < N) ? nIdx : (N - 1);
    const float* bPtr = item_emb + (size_t)nClamped * K_DIM + 2 * half;

    v8f acc = {};
#pragma unroll
    for (int t = 0; t < K_DIM / 4; ++t) {
      v2f a = *(const v2f*)(aPtr + 4 * t);
      v2f b = *(const v2f*)(bPtr + 4 * t);
      // (neg_a, A, neg_b, B, c_mod, C, reuse_a, reuse_b)
      acc = __builtin_amdgcn_wmma_f32_16x16x4_f32(
          false, a, false, b, (short)0, acc, false, false);
    }

    // D layout: VGPR v, lane l -> row (v + 8*half), col ml
    if (nIdx < N) {
      float* outBase = pref + (size_t)(mTile * TILE + 8 * half) * N
                            + (size_t)nTile * TILE + ml;
#pragma unroll
      for (int v = 0; v < 8; ++v) {
        float s = acc[v];
        outBase[(size_t)v * N] = s;
        lsum += s * s;
      }
    }
  }

  // wave32 reduction of sum(s^2)
#pragma unroll
  for (int off = 16; off > 0; off >>= 1)
    lsum += __shfl_xor(lsum, off, 32);
  if (lane == 0) red[waveId] = lsum;
  __syncthreads();
  if (threadIdx.x == 0) {
    float s = 0.0f;
    const int nw = blockDim.x >> 5;
    for (int w = 0; w < nw; ++w) s += red[w];
    atomicAdd(loss, s);
  }
}

// Positives: loss += (s-1)^2 - s^2 = 1 - 2s per UNIQUE (row, index) pair,
// matching the reference's boolean-scatter (set) semantics.
__global__ __launch_bounds__(64) void mf_pos_loss(
    const int*   __restrict__ seq,
    const float* __restrict__ pref,
    float* __restrict__ loss,
    int HIST, int N)
{
  __shared__ float red[64];
  const int b = blockIdx.x;
  const int j = threadIdx.x;
  float c = 0.0f;
  if (j < HIST) {
    const int idx = seq[b * HIST + j];
    bool dup = false;
    for (int p = 0; p < j; ++p) dup |= (seq[b * HIST + p] == idx);
    if (!dup) {
      const float s = pref[(size_t)b * N + idx];
      c = 1.0f - 2.0f * s;
    }
  }
  red[j] = c;
  __syncthreads();
#pragma unroll
  for (int off = 32; off > 0; off >>= 1) {
    if (j < off) red[j] += red[j + off];
    __syncthreads();
  }
  if (j == 0) atomicAdd(loss, red[0]);
}

extern "C" void kernel_launch(void* const* d_in, const int* in_sizes, int n_in,
                              void* d_out, int out_size, void* d_ws, size_t ws_size,
                              hipStream_t stream) {
  const float* user_emb   = (const float*)d_in[0];
  const float* item_emb   = (const float*)d_in[1];
  const int*   center_uid = (const int*)d_in[2];
  const int*   seq        = (const int*)d_in[3];

  const int B    = in_sizes[2];            // 1024
  const int HIST = in_sizes[3] / B;        // 50
  const int N    = in_sizes[1] / K_DIM;    // 100000

  float* loss = (float*)d_out;             // d_out[0] = loss
  float* pref = loss + 1;                  // d_out[1..] = pref [B, N]

  hipMemsetAsync(d_out, 0, sizeof(float), stream);

  const int MT = (B + TILE - 1) / TILE;    // 64
  const int NT = (N + TILE - 1) / TILE;    // 6250
  const int totalTiles = MT * NT;          // 400000
  const int wavesPerBlock = 8;             // 256 threads
  const int blocks = (totalTiles + wavesPerBlock - 1) / wavesPerBlock;

  mf_gemm_wmma<<<blocks, 256, 0, stream>>>(user_emb, item_emb, center_uid,
                                           pref, loss, B, N, NT, totalTiles);
  mf_pos_loss<<<B, 64, 0, stream>>>(seq, pref, loss, HIST, N);
}